// SharedNReadChain_64123861729640
// MI455X (gfx1250) — compile-verified
//
#include <hip/hip_runtime.h>
#include <math.h>

// Problem constants (match reference)
#define B_ 128
#define S_ 2048
#define D_ 512
#define E_ 50000
#define H_ 5
#define EPS_ 1e-12f
#define KC 64   // K-chunk staged in LDS for the WMMA GEMM

typedef __attribute__((ext_vector_type(2))) float v2f;
typedef __attribute__((ext_vector_type(4))) float v4f;
typedef __attribute__((ext_vector_type(8))) float v8f;

// ---------------- wave helpers (wave32) ----------------
__device__ __forceinline__ float wave_reduce_sum(float v) {
    #pragma unroll
    for (int off = 16; off > 0; off >>= 1)
        v += __shfl_xor(v, off, 32);
    return v;
}

// ---------------- 1/||keys[b,s,:]|| (hop-invariant, computed once) ----------------
__global__ void keynorm_kernel(const float* __restrict__ keys, float* __restrict__ invk) {
    const int wave = threadIdx.x >> 5, lane = threadIdx.x & 31;
    const long gw = (long)blockIdx.x * 8 + wave;          // flat (b*S + s)
    const v4f* kp4 = (const v4f*)(keys + gw * D_);        // rows are 2KB aligned
    float ss = 0.f;
    #pragma unroll
    for (int i = 0; i < D_ / 128; ++i) {
        const v4f x = __builtin_nontemporal_load(kp4 + lane + 32 * i);  // NT: protect L2 for We
        ss += x[0]*x[0] + x[1]*x[1] + x[2]*x[2] + x[3]*x[3];
    }
    ss = wave_reduce_sum(ss);
    if (lane == 0) invk[gw] = 1.0f / fmaxf(sqrtf(ss), EPS_);
}

// ---------------- state += hop_emb[hop] ----------------
__global__ void addemb_kernel(float* __restrict__ state, const float* __restrict__ hop_emb, int hop) {
    const int i = blockIdx.x * blockDim.x + threadIdx.x;   // < B*D
    state[i] += hop_emb[hop * D_ + (i & (D_ - 1))];
}

// -------- out[128 x N] = A[128 x K] @ W[N x K]^T + bias  via V_WMMA_F32_16X16X4_F32 --------
// Workgroup: 8 waves; wave w owns M rows [16w,16w+16) x (16*NT) output cols.
// NT n-tiles per WG amortize the A-panel staging (L2 traffic / NT) and reuse A frags NT times.
// Software-pipelined: chunk c+1 global loads issued while chunk c's WMMA bursts execute.
template <int NT>
__global__ void gemm_bt_wmma_kernel(const float* __restrict__ A,
                                    const float* __restrict__ W,
                                    const float* __restrict__ bias,
                                    float* __restrict__ out,
                                    int N, int K) {
    __shared__ float lds_a[128][KC + 1];        // +1 pad: conflict-free column reads
    __shared__ float lds_w[16 * NT][KC + 1];

    const int tid  = threadIdx.x;
    const int wave = tid >> 5, lane = tid & 31;
    const int half = lane >> 4, l = lane & 15;  // C/D layout: lanes 0-15 -> M=i, 16-31 -> M=i+8
    const int n0 = blockIdx.x * (16 * NT);
    const int m0 = wave * 16;

    v8f acc[NT];
    #pragma unroll
    for (int j = 0; j < NT; ++j) {
        const float bval = bias[n0 + 16 * j + l];   // bias depends only on N column
        #pragma unroll
        for (int i = 0; i < 8; ++i) acc[j][i] = bval;
    }

    const int nchunks = K / KC;

    // ---- prologue: global loads for chunk 0 ----
    v4f wreg[NT];
    #pragma unroll
    for (int j = 0; j < NT; ++j) {
        const int idx = tid + (j << 8);
        const int r = idx >> 4, c4 = (idx & 15) << 2;
        wreg[j] = *(const v4f*)(W + (long)(n0 + r) * K + c4);
    }
    v4f areg[8];
    #pragma unroll
    for (int i = 0; i < 8; ++i) {
        const int idx = tid + (i << 8);
        const int r = idx >> 4, c4 = (idx & 15) << 2;
        areg[i] = *(const v4f*)(A + (long)r * K + c4);
    }

    for (int ch = 0; ch < nchunks; ++ch) {
        __syncthreads();   // previous chunk's LDS reads complete
        // stage registers -> LDS
        #pragma unroll
        for (int j = 0; j < NT; ++j) {
            const int idx = tid + (j << 8);
            const int r = idx >> 4, c4 = (idx & 15) << 2;
            lds_w[r][c4 + 0] = wreg[j][0]; lds_w[r][c4 + 1] = wreg[j][1];
            lds_w[r][c4 + 2] = wreg[j][2]; lds_w[r][c4 + 3] = wreg[j][3];
        }
        #pragma unroll
        for (int i = 0; i < 8; ++i) {
            const int idx = tid + (i << 8);
            const int r = idx >> 4, c4 = (idx & 15) << 2;
            lds_a[r][c4 + 0] = areg[i][0]; lds_a[r][c4 + 1] = areg[i][1];
            lds_a[r][c4 + 2] = areg[i][2]; lds_a[r][c4 + 3] = areg[i][3];
        }
        __syncthreads();

        // ---- issue next chunk's global loads (overlap with WMMA below) ----
        if (ch + 1 < nchunks) {
            const int kc0 = (ch + 1) * KC;
            #pragma unroll
            for (int j = 0; j < NT; ++j) {
                const int idx = tid + (j << 8);
                const int r = idx >> 4, c4 = (idx & 15) << 2;
                wreg[j] = *(const v4f*)(W + (long)(n0 + r) * K + kc0 + c4);
            }
            #pragma unroll
            for (int i = 0; i < 8; ++i) {
                const int idx = tid + (i << 8);
                const int r = idx >> 4, c4 = (idx & 15) << 2;
                areg[i] = *(const v4f*)(A + (long)r * K + kc0 + c4);
            }
        }

        // ---- 16 WMMA K-steps x NT tiles: A frags prefetched once, reused NT times ----
        #pragma unroll
        for (int kh = 0; kh < 2; ++kh) {
            v2f af[8];
            #pragma unroll
            for (int kk = 0; kk < 8; ++kk) {
                // f32 A 16x4 layout: lanes 0-15 hold {K=0,K=1}, lanes 16-31 hold {K=2,K=3}
                const int ka = (((kh << 3) + kk) << 2) + (half << 1);
                af[kk][0] = lds_a[m0 + l][ka]; af[kk][1] = lds_a[m0 + l][ka + 1];
            }
            #pragma unroll
            for (int j = 0; j < NT; ++j) {
                v2f wf[8];
                #pragma unroll
                for (int kk = 0; kk < 8; ++kk) {
                    const int ka = (((kh << 3) + kk) << 2) + (half << 1);
                    wf[kk][0] = lds_w[16 * j + l][ka];
                    wf[kk][1] = lds_w[16 * j + l][ka + 1];
                }
                #pragma unroll
                for (int kk = 0; kk < 8; ++kk)
                    acc[j] = __builtin_amdgcn_wmma_f32_16x16x4_f32(false, af[kk], false, wf[kk],
                                                                   (short)0, acc[j], false, false);
            }
        }
    }
    // D layout: VGPR i -> M = m0 + i + 8*half, N = n0 + 16*j + l
    #pragma unroll
    for (int j = 0; j < NT; ++j) {
        #pragma unroll
        for (int i = 0; i < 8; ++i) {
            const int m = m0 + i + (half << 3);
            out[(long)m * N + (n0 + 16 * j + l)] = acc[j][i];
        }
    }
}

// ---------------- L2-normalize rows of q in place ----------------
__global__ void rownorm_kernel(float* __restrict__ q) {
    const int b = blockIdx.x;
    float* row = q + (long)b * D_;
    __shared__ float red[8];
    const int lane = threadIdx.x & 31, wave = threadIdx.x >> 5;
    float ss = 0.f;
    #pragma unroll
    for (int i = 0; i < D_ / 256; ++i) {
        float x = row[threadIdx.x + 256 * i];
        ss += x * x;
    }
    ss = wave_reduce_sum(ss);
    if (lane == 0) red[wave] = ss;
    __syncthreads();
    float tot = 0.f;
    #pragma unroll
    for (int i = 0; i < 8; ++i) tot += red[i];
    const float inv = 1.0f / fmaxf(sqrtf(tot), EPS_);
    #pragma unroll
    for (int i = 0; i < D_ / 256; ++i) row[threadIdx.x + 256 * i] *= inv;
}

// ---------------- attn_logits[b,s] = dot(q_norm[b], keys[b,s]) * invk[b,s] ----------------
__global__ void dots_kernel(const float* __restrict__ qn, const float* __restrict__ keys,
                            const float* __restrict__ invk, float* __restrict__ logits) {
    const int wave = threadIdx.x >> 5, lane = threadIdx.x & 31;
    const long gw = (long)blockIdx.x * 8 + wave;          // (b*S + s)
    const int b = (int)(gw >> 11);                        // / S_
    const v4f* kp4 = (const v4f*)(keys + gw * D_);
    const v4f* qp4 = (const v4f*)(qn + (long)b * D_);
    float s = 0.f;
    #pragma unroll
    for (int i = 0; i < D_ / 128; ++i) {
        const v4f xq = qp4[lane + 32 * i];
        const v4f xk = __builtin_nontemporal_load(kp4 + lane + 32 * i);
        s += xq[0]*xk[0] + xq[1]*xk[1] + xq[2]*xk[2] + xq[3]*xk[3];
    }
    s = wave_reduce_sum(s);
    if (lane == 0) logits[gw] = s * invk[gw];
}

// ---------------- softmax over S per (hop,b); writes attn to d_out and back to ws ----------------
__global__ void softmax_kernel(float* __restrict__ logits, float* __restrict__ attn_out) {
    const int b = blockIdx.x;
    float* lp = logits + (long)b * S_;
    __shared__ float red[256];
    float v[S_ / 256];
    float mx = -INFINITY;
    #pragma unroll
    for (int i = 0; i < S_ / 256; ++i) { v[i] = lp[threadIdx.x + 256 * i]; mx = fmaxf(mx, v[i]); }
    red[threadIdx.x] = mx; __syncthreads();
    for (int st = 128; st > 0; st >>= 1) {
        if (threadIdx.x < st) red[threadIdx.x] = fmaxf(red[threadIdx.x], red[threadIdx.x + st]);
        __syncthreads();
    }
    mx = red[0]; __syncthreads();
    float sum = 0.f;
    #pragma unroll
    for (int i = 0; i < S_ / 256; ++i) { v[i] = expf(v[i] - mx); sum += v[i]; }
    red[threadIdx.x] = sum; __syncthreads();
    for (int st = 128; st > 0; st >>= 1) {
        if (threadIdx.x < st) red[threadIdx.x] += red[threadIdx.x + st];
        __syncthreads();
    }
    const float inv = 1.0f / red[0];
    #pragma unroll
    for (int i = 0; i < S_ / 256; ++i) {
        const float a = v[i] * inv;
        lp[threadIdx.x + 256 * i] = a;
        attn_out[(long)b * S_ + threadIdx.x + 256 * i] = a;
    }
}

// ---------------- read[b,d] = sum_s attn[b,s] * values[b,s,d]  (S chunked for occupancy) ----------
__global__ void read_kernel(const float* __restrict__ attn, const float* __restrict__ values,
                            float* __restrict__ rbuf) {
    const int b  = blockIdx.x;
    const int s0 = blockIdx.y * (S_ / 8);
    const int t  = threadIdx.x;                 // covers d pairs: d = 2*t, 2*t+1
    const float* ap = attn + (long)b * S_ + s0;
    const v2f* vp2  = (const v2f*)(values + ((long)b * S_ + s0) * D_) + t;
    v2f acc; acc[0] = 0.f; acc[1] = 0.f;
    #pragma unroll 4
    for (int s = 0; s < S_ / 8; ++s) {
        const float a = ap[s];
        const v2f v = __builtin_nontemporal_load(vp2 + (long)s * (D_ / 2));  // NT streaming
        acc[0] += a * v[0];
        acc[1] += a * v[1];
    }
    atomicAdd(&rbuf[b * D_ + 2 * t + 0], acc[0]);
    atomicAdd(&rbuf[b * D_ + 2 * t + 1], acc[1]);
}

// ---------------- state += read ----------------
__global__ void addread_kernel(float* __restrict__ state, const float* __restrict__ rbuf) {
    const int i = blockIdx.x * blockDim.x + threadIdx.x;
    state[i] += rbuf[i];
}

extern "C" void kernel_launch(void* const* d_in, const int* in_sizes, int n_in,
                              void* d_out, int out_size, void* d_ws, size_t ws_size,
                              hipStream_t stream) {
    (void)in_sizes; (void)n_in; (void)out_size; (void)ws_size;
    const float* query   = (const float*)d_in[0];
    const float* keys    = (const float*)d_in[1];
    const float* values  = (const float*)d_in[2];
    const float* Wq      = (const float*)d_in[3];
    const float* bq      = (const float*)d_in[4];
    const float* We      = (const float*)d_in[5];
    const float* be      = (const float*)d_in[6];
    const float* hop_emb = (const float*)d_in[7];

    float* out        = (float*)d_out;
    float* out_state  = out;                                      // [B,D]
    float* out_logits = out + (long)B_ * D_;                      // [H,B,E]
    float* out_attn   = out + (long)B_ * D_ + (long)H_ * B_ * E_; // [H,B,S]

    float* ws    = (float*)d_ws;
    float* state = ws;                          // B*D
    float* q     = state + B_ * D_;             // B*D
    float* invk  = q + B_ * D_;                 // B*S
    float* slog  = invk + (long)B_ * S_;        // B*S (logits, then attn in place)
    float* rbuf  = slog + (long)B_ * S_;        // B*D

    hipMemcpyAsync(state, query, (size_t)B_ * D_ * sizeof(float),
                   hipMemcpyDeviceToDevice, stream);
    keynorm_kernel<<<B_ * S_ / 8, 256, 0, stream>>>(keys, invk);

    for (int hop = 0; hop < H_; ++hop) {
        addemb_kernel<<<B_ * D_ / 256, 256, 0, stream>>>(state, hop_emb, hop);
        gemm_bt_wmma_kernel<1><<<D_ / 16, 256, 0, stream>>>(state, Wq, bq, q, D_, D_);
        rownorm_kernel<<<B_, 256, 0, stream>>>(q);
        dots_kernel<<<B_ * S_ / 8, 256, 0, stream>>>(q, keys, invk, slog);
        softmax_kernel<<<B_, 256, 0, stream>>>(slog, out_attn + (long)hop * B_ * S_);
        hipMemsetAsync(rbuf, 0, (size_t)B_ * D_ * sizeof(float), stream);
        read_kernel<<<dim3(B_, 8), 256, 0, stream>>>(slog, values, rbuf);
        addread_kernel<<<B_ * D_ / 256, 256, 0, stream>>>(state, rbuf);
        gemm_bt_wmma_kernel<5><<<E_ / 80, 256, 0, stream>>>(state, We, be,
                                                            out_logits + (long)hop * B_ * E_,
                                                            E_, D_);
    }
    hipMemcpyAsync(out_state, state, (size_t)B_ * D_ * sizeof(float),
                   hipMemcpyDeviceToDevice, stream);
}